// LabelSmoothing_25778393710899
// MI455X (gfx1250) — compile-verified
//
#include <hip/hip_runtime.h>
#include <hip/hip_bf16.h>
#include <cmath>

typedef float v2f __attribute__((ext_vector_type(2)));
typedef float v8f __attribute__((ext_vector_type(8)));
typedef float f4  __attribute__((ext_vector_type(4)));

// Full 32-lane wave sum via V_WMMA_F32_16X16X4_F32 with B = ones:
//   A: vgpr0 = acc (lanes 0-15 -> A[m,0], lanes 16-31 -> A[m,2]), vgpr1 = 0
//   D[m,n] = sum_k A[m,k] = acc[m] + acc[m+16]   (same for every column n)
// Summing the 8 D vgprs gives half-sums per lane half; one xor-16 shuffle
// finishes. Callers guarantee EXEC == all ones (full blocks, no divergence).
__device__ __forceinline__ float wave_sum32(float acc) {
#if defined(__gfx1250__) && __has_builtin(__builtin_amdgcn_wmma_f32_16x16x4_f32)
  v2f a; a.x = acc;  a.y = 0.0f;
  v2f b; b.x = 1.0f; b.y = 1.0f;
  v8f c = {};
  c = __builtin_amdgcn_wmma_f32_16x16x4_f32(false, a, false, b, (short)0, c,
                                            false, false);
  float s = ((c[0] + c[1]) + (c[2] + c[3])) + ((c[4] + c[5]) + (c[6] + c[7]));
  s += __shfl_xor(s, 16, 32);
  return s;
#else
  for (int off = 16; off > 0; off >>= 1) acc += __shfl_xor(acc, off, 32);
  return acc;
#endif
}

__device__ __forceinline__ float hsum4(f4 v) {
  return (v.x + v.y) + (v.z + v.w);
}

// One block per row: streaming masked row-sum + per-row gather terms.
__global__ __launch_bounds__(256)
void ls_row_kernel(const float* __restrict__ x, const int* __restrict__ tgt,
                   float* __restrict__ partials, int V,
                   float eps, float conf, float rowc) {
  const int row = blockIdx.x;
  const float* xr = x + (size_t)row * (size_t)V;
  const int t = tgt[row];
  const int valid = (t != 0);

  const f4* xv = (const f4*)xr;            // 16B-aligned for V % 4 == 0
  const int nvec = V >> 2;
  const int stride = blockDim.x;           // 256

  // 4x unroll, 4 independent accumulators: 4 global_load_b128 in flight per
  // thread per iteration before the first s_wait_loadcnt; breaks the FP-add
  // dependency chain so the wave can stream at HBM rate.
  float a0 = 0.0f, a1 = 0.0f, a2 = 0.0f, a3 = 0.0f;
  int i = threadIdx.x;
  for (; i + 3 * stride < nvec; i += 4 * stride) {
    f4 v0 = __builtin_nontemporal_load(&xv[i]);
    f4 v1 = __builtin_nontemporal_load(&xv[i + stride]);
    f4 v2 = __builtin_nontemporal_load(&xv[i + 2 * stride]);
    f4 v3 = __builtin_nontemporal_load(&xv[i + 3 * stride]);
    a0 += hsum4(v0);
    a1 += hsum4(v1);
    a2 += hsum4(v2);
    a3 += hsum4(v3);
  }
  for (; i < nvec; i += stride) {
    a0 += hsum4(__builtin_nontemporal_load(&xv[i]));
  }
  // Scalar tail (empty for V = 32000).
  for (int j = (nvec << 2) + threadIdx.x; j < V; j += stride) a1 += xr[j];

  float acc = (a0 + a1) + (a2 + a3);
  acc = valid ? acc : 0.0f;                // uniform -> cndmask, EXEC intact

  float w = wave_sum32(acc);               // WMMA-based 32-lane reduction

  __shared__ float lds[8];
  const int lane = threadIdx.x & 31;
  const int wid  = threadIdx.x >> 5;
  if (lane == 0) lds[wid] = w;
  __syncthreads();

  if (threadIdx.x == 0) {
    float s = 0.0f;
    const int nw = blockDim.x >> 5;
    for (int k = 0; k < nw; ++k) s += lds[k];
    // loss_n = C_row - (conf-eps)*x[n,t] + eps*x[n,0] - eps*rowsum  (0 if pad)
    float g = valid ? (rowc - (conf - eps) * xr[t] + eps * xr[0]) : 0.0f;
    partials[row] = g - eps * s;
  }
}

// Single-block deterministic reduction of the per-row partials.
__global__ __launch_bounds__(1024)
void ls_final_kernel(const float* __restrict__ partials, int n,
                     float* __restrict__ out) {
  float acc = 0.0f;
  for (int i = threadIdx.x; i < n; i += blockDim.x) acc += partials[i];

  float w = wave_sum32(acc);

  __shared__ float lds[32];
  const int lane = threadIdx.x & 31;
  const int wid  = threadIdx.x >> 5;
  if (lane == 0) lds[wid] = w;
  __syncthreads();

  if (threadIdx.x == 0) {
    float s = 0.0f;
    const int nw = blockDim.x >> 5;
    for (int k = 0; k < nw; ++k) s += lds[k];
    out[0] = s;
  }
}

extern "C" void kernel_launch(void* const* d_in, const int* in_sizes, int n_in,
                              void* d_out, int out_size, void* d_ws, size_t ws_size,
                              hipStream_t stream) {
  const float* x   = (const float*)d_in[0];   // (N, V) fp32 log-probs
  const int*   tgt = (const int*)d_in[1];     // (N,) targets
  float*       out = (float*)d_out;           // scalar fp32

  const int n = in_sizes[1];
  const int V = in_sizes[0] / n;

  float* partials = (float*)d_ws;             // N floats of scratch

  // Match jnp's f32 evaluation: eps rounded to f32, logs of the f32 values.
  const float  epsf  = (float)(0.1 / (double)(V - 2));
  const float  conff = 0.9f;
  const float  rowc  = (float)((double)conff * log((double)conff) +
                               (double)(V - 2) * (double)epsf * log((double)epsf));

  ls_row_kernel<<<n, 256, 0, stream>>>(x, tgt, partials, V, epsf, conff, rowc);
  ls_final_kernel<<<1, 1024, 0, stream>>>(partials, n, out);
}